// LGA_89395449299015
// MI455X (gfx1250) — compile-verified
//
#include <hip/hip_runtime.h>
#include <stdint.h>

#define AS1 __attribute__((address_space(1)))
#define AS3 __attribute__((address_space(3)))

typedef __attribute__((ext_vector_type(2))) float v2f;
typedef __attribute__((ext_vector_type(2))) int   v2i;

namespace {
constexpr int N_ = 4, C_ = 32, H_ = 384, W_ = 768;
constexpr int RAD = 2, KD = 5, KK = 25;
constexpr int TH = 8, TW = 64;          // output tile per block
constexpr int LR = TH + 2 * RAD;        // 12 LDS rows (with halo)
constexpr int LC = TW + 2 * RAD;        // 68 LDS cols (with halo)
constexpr int PAIRS = LR * LC / 2;      // 408 float2 slots per buffer
constexpr int NTHREADS = 256;
}

#if defined(__has_builtin)
#if __has_builtin(__builtin_amdgcn_global_load_async_to_lds_b64)
#define LGA_ASYNC 1
#endif
#endif

__device__ __forceinline__ void lga_wait_async() {
#if defined(LGA_ASYNC)
#if __has_builtin(__builtin_amdgcn_s_wait_asynccnt)
  __builtin_amdgcn_s_wait_asynccnt(0);
#else
  asm volatile("s_wait_asynccnt 0" ::: "memory");
#endif
#endif
}

__global__ __launch_bounds__(NTHREADS) void lga_kernel(
    const float* __restrict__ in1,   // [N,C,H,W]
    const float* __restrict__ in2,   // [N,25,H,W]
    float* __restrict__ out)         // [N,C,H,W]
{
  __shared__ __align__(16) float tile[2][LR * LC];  // double buffer, 6528 B

  const int tid = threadIdx.x;
  const int txx = tid & 31;          // lane within wave = tile col group
  const int tyy = tid >> 5;          // wave id = tile row
  const int n  = blockIdx.z;
  const int h0 = blockIdx.y * TH;
  const int w0 = blockIdx.x * TW;
  const int p  = txx * 2;            // this thread's output col pair
  const int q  = tyy;                // this thread's output row

  // ---- staging descriptors: 2 float2 halo slots per thread ----------------
  int lofs[2]; int gofs[2]; bool inb[2]; bool valid[2];
#pragma unroll
  for (int k = 0; k < 2; ++k) {
    const int pidx = tid + k * NTHREADS;
    valid[k] = (pidx < PAIRS);
    const int row = pidx / (LC / 2);
    const int col = (pidx - row * (LC / 2)) * 2;
    const int gh = h0 + row - RAD;
    const int gw = w0 + col - RAD;
    // pair is entirely in or entirely out of bounds (even col start)
    inb[k]  = valid[k] && gh >= 0 && gh < H_ && gw >= 0 && (gw + 1) < W_;
    lofs[k] = row * LC + col;
    gofs[k] = gh * W_ + gw;
  }

  // Zero-fill OOB halo slots once per buffer: async loads never touch them,
  // so zeros persist across the whole channel loop (zero padding semantics).
#pragma unroll
  for (int k = 0; k < 2; ++k) {
    if (valid[k] && !inb[k]) {
      *(v2f*)&tile[0][lofs[k]] = (v2f)0.f;
      *(v2f*)&tile[1][lofs[k]] = (v2f)0.f;
    }
  }

  // ---- per-pixel 5x5 guide weights: channel-invariant -> keep in VGPRs ----
  v2f wt[KK];
  {
    const float* wsrc = in2 + ((n * KK) * H_ + (h0 + q)) * W_ + (w0 + p);
#pragma unroll
    for (int t = 0; t < KK; ++t)
      wt[t] = __builtin_nontemporal_load((const v2f*)(wsrc + t * (H_ * W_)));
  }

  const float* src0 = in1 + (n * C_) * (H_ * W_);

  auto stage = [&](int c, int buf) {
    const float* src = src0 + c * (H_ * W_);
#pragma unroll
    for (int k = 0; k < 2; ++k) {
      if (inb[k]) {
#if defined(LGA_ASYNC)
        __builtin_amdgcn_global_load_async_to_lds_b64(
            (AS1 v2i*)(src + gofs[k]),
            (AS3 v2i*)&tile[buf][lofs[k]], 0, 0);
#else
        *(v2f*)&tile[buf][lofs[k]] = *(const v2f*)(src + gofs[k]);
#endif
      }
    }
  };

  stage(0, 0);  // prefetch channel 0 into buffer 0

  float* dst = out + ((n * C_) * H_ + (h0 + q)) * W_ + (w0 + p);

  for (int c = 0; c < C_; ++c) {
    lga_wait_async();      // this wave's asyncs for buffer (c&1) have landed
    __syncthreads();       // all waves' fills visible; all reads of (c+1)&1
                           // buffer from iteration c-1 are complete
    if (c + 1 < C_) stage(c + 1, (c + 1) & 1);  // overlap with compute below

    const float* tb = tile[c & 1];
    v2f acc = (v2f)0.f;
#pragma unroll
    for (int i = 0; i < KD; ++i) {
      const float* rowp = tb + (q + i) * LC + p;   // 8B-aligned, stride-2 lanes
      v2f f0 = *(const v2f*)(rowp + 0);            // cols p..p+1
      v2f f1 = *(const v2f*)(rowp + 2);            // cols p+2..p+3
      v2f f2 = *(const v2f*)(rowp + 4);            // cols p+4..p+5
      v2f cc1 = {f0.y, f1.x};
      v2f cc3 = {f1.y, f2.x};
      acc += f0  * wt[i * KD + 0];
      acc += cc1 * wt[i * KD + 1];
      acc += f1  * wt[i * KD + 2];
      acc += cc3 * wt[i * KD + 3];
      acc += f2  * wt[i * KD + 4];
    }
    __builtin_nontemporal_store(acc, (v2f*)(dst + c * (H_ * W_)));
  }
}

extern "C" void kernel_launch(void* const* d_in, const int* in_sizes, int n_in,
                              void* d_out, int out_size, void* d_ws, size_t ws_size,
                              hipStream_t stream) {
  const float* in1 = (const float*)d_in[0];
  const float* in2 = (const float*)d_in[1];
  float* outp = (float*)d_out;
  dim3 grid(W_ / TW, H_ / TH, N_);   // 12 x 48 x 4 = 2304 blocks
  dim3 block(NTHREADS);
  hipLaunchKernelGGL(lga_kernel, grid, block, 0, stream, in1, in2, outp);
}